// MessagingLayer_4964982194953
// MI455X (gfx1250) — compile-verified
//
#include <hip/hip_runtime.h>
#include <math.h>

typedef float v2f __attribute__((ext_vector_type(2)));
typedef float v8f __attribute__((ext_vector_type(8)));

#define N_NODES      100000
#define DIM          256
#define N_EDGE_TYPES 6
#define M_PER_TYPE   400000
#define POS_TABLE    512
#define SMALL_NUMBER 1e-8f

// ---------------------------------------------------------------------------
// K0: zero the output accumulator and the bincount (must happen every call).
// ---------------------------------------------------------------------------
__global__ void k_zero(float* __restrict__ out, int* __restrict__ bincount) {
    long long stride = (long long)gridDim.x * blockDim.x;
    long long i = (long long)blockIdx.x * blockDim.x + threadIdx.x;
    const long long total = (long long)N_NODES * DIM;
    for (long long j = i; j < total; j += stride) out[j] = 0.0f;
    for (long long j = i; j < N_NODES; j += stride) bincount[j] = 0;
}

// ---------------------------------------------------------------------------
// K1: pos_gating[p, d] = 2 * sigmoid( pos_emb[p, :] . Wp[d, :] + bp[d] )
// pos_emb computed on the fly into LDS (sin half / cos half / 2 zero pad).
// One block per position, 256 threads (one output column each).
// ---------------------------------------------------------------------------
__global__ void k_gating(const float* __restrict__ Wp,
                         const float* __restrict__ bp,
                         float* __restrict__ gating) {
    __shared__ float pe[DIM];
    const int p = blockIdx.x;      // 0..511
    const int t = threadIdx.x;     // 0..255
    const float log1e4 = logf(10000.0f);
    float v;
    if (t < 127) {
        float invf = expf(-((float)(2 * t) / 254.0f) * log1e4);
        v = sinf((float)p * invf);
    } else if (t < 254) {
        int j = t - 127;
        float invf = expf(-((float)(2 * j) / 254.0f) * log1e4);
        v = cosf((float)p * invf);
    } else {
        v = 0.0f;                  // 2-column selector pad
    }
    pe[t] = v;
    __syncthreads();

    float acc = bp[t];
    const float* wrow = Wp + (long long)t * DIM;   // Wp[d, k], row-major
    #pragma unroll 8
    for (int k = 0; k < DIM; ++k) acc += pe[k] * wrow[k];
    gating[(long long)p * DIM + t] = 2.0f / (1.0f + expf(-acc));
}

// ---------------------------------------------------------------------------
// K2: prop_i = node_states @ W_i^T + b_i   (M=100000, N=256, K=256), fp32 WMMA.
// W_i = W[etype*256 : (etype+1)*256, :], both operands row-major with K
// contiguous, so A and B lane loads share the same float2 addressing pattern
// per the 16x4 / 4x16 f32 VGPR layout (CDNA5 ISA §7.12.2).
// Block: 256 threads = 8 waves, arranged 2(M) x 4(N); wave tile = 16x64.
// ---------------------------------------------------------------------------
__global__ void k_gemm(const float* __restrict__ ns,
                       const float* __restrict__ W,
                       const float* __restrict__ b,
                       float* __restrict__ prop,
                       int etype) {
    const int lane  = threadIdx.x & 31;
    const int wave  = threadIdx.x >> 5;     // 0..7
    const int wm    = wave & 1;             // 2 waves along M
    const int wn    = wave >> 1;            // 4 waves along N
    const int m0    = blockIdx.x * 32 + wm * 16;
    const int n0    = wn * 64;
    const int l15   = lane & 15;
    const int lhalf = lane >> 4;            // 0: K pair {0,1}, 1: K pair {2,3}

    const float* arow  = ns + (long long)(m0 + l15) * DIM;
    const float* Wbase = W + (long long)etype * DIM * DIM;
    const int    ncol  = n0 + l15;
    const float* brow0 = Wbase + (long long)(ncol +  0) * DIM;
    const float* brow1 = Wbase + (long long)(ncol + 16) * DIM;
    const float* brow2 = Wbase + (long long)(ncol + 32) * DIM;
    const float* brow3 = Wbase + (long long)(ncol + 48) * DIM;

    v8f acc0 = {}, acc1 = {}, acc2 = {}, acc3 = {};

    #pragma unroll 4
    for (int k0 = 0; k0 < DIM; k0 += 4) {
        const int kk = k0 + 2 * lhalf;
        v2f a  = *(const v2f*)(arow  + kk);
        v2f b0 = *(const v2f*)(brow0 + kk);
        v2f b1 = *(const v2f*)(brow1 + kk);
        v2f b2 = *(const v2f*)(brow2 + kk);
        v2f b3 = *(const v2f*)(brow3 + kk);
        acc0 = __builtin_amdgcn_wmma_f32_16x16x4_f32(false, a, false, b0, (short)0, acc0, false, false);
        acc1 = __builtin_amdgcn_wmma_f32_16x16x4_f32(false, a, false, b1, (short)0, acc1, false, false);
        acc2 = __builtin_amdgcn_wmma_f32_16x16x4_f32(false, a, false, b2, (short)0, acc2, false, false);
        acc3 = __builtin_amdgcn_wmma_f32_16x16x4_f32(false, a, false, b3, (short)0, acc3, false, false);
    }

    // Epilogue: add bias, store. C layout: lanes 0-15 -> M = v, lanes 16-31 ->
    // M = 8 + v; N = lane & 15 (per 16x16 f32 C/D layout).
    const float* bvec = b + (long long)etype * DIM;
    const int row_base = m0 + 8 * lhalf;
    v8f accs[4] = {acc0, acc1, acc2, acc3};
    #pragma unroll
    for (int t = 0; t < 4; ++t) {
        const int col = n0 + t * 16 + l15;
        const float bias = bvec[col];
        #pragma unroll
        for (int v = 0; v < 8; ++v) {
            prop[(long long)(row_base + v) * DIM + col] = accs[t][v] + bias;
        }
    }
}

// ---------------------------------------------------------------------------
// K3: scatter-accumulate one edge type. 64 threads per edge; each thread
// handles 4 contiguous floats (float4 gather of prop & gating rows), then
// 4 global f32 atomic adds into out[tgt]. Lane 0 bumps bincount[tgt].
// ---------------------------------------------------------------------------
__global__ void k_scatter(const int* __restrict__ edges,   // [M_PER_TYPE, 2]
                          const int* __restrict__ pos,     // [M_PER_TYPE]
                          const float* __restrict__ prop,  // [N_NODES, DIM]
                          const float* __restrict__ gating,// [POS_TABLE, DIM]
                          float* __restrict__ out,
                          int* __restrict__ bincount) {
    const int t      = threadIdx.x;
    const int esub   = t >> 6;              // 4 edges per 256-thread block
    const int lane64 = t & 63;
    const long long e = (long long)blockIdx.x * 4 + esub;
    if (e >= M_PER_TYPE) return;

    const int src = edges[e * 2 + 0];
    const int tgt = edges[e * 2 + 1];
    const int p   = pos[e];
    const int d   = lane64 * 4;

    const float4 pv = *(const float4*)(prop   + (long long)src * DIM + d);
    const float4 gv = *(const float4*)(gating + (long long)p   * DIM + d);
    float* o = out + (long long)tgt * DIM + d;
    atomicAdd(o + 0, pv.x * gv.x);
    atomicAdd(o + 1, pv.y * gv.y);
    atomicAdd(o + 2, pv.z * gv.z);
    atomicAdd(o + 3, pv.w * gv.w);
    if (lane64 == 0) atomicAdd(bincount + tgt, 1);
}

// ---------------------------------------------------------------------------
// K4: out /= (divisor + eps), divisor = (bincount == 0) ? 1 : bincount
// ---------------------------------------------------------------------------
__global__ void k_div(float* __restrict__ out, const int* __restrict__ bincount) {
    const long long i = (long long)blockIdx.x * blockDim.x + threadIdx.x;
    if (i >= (long long)N_NODES * DIM) return;
    const int n  = (int)(i >> 8);           // DIM == 256
    const int bc = bincount[n];
    const float div = ((bc == 0) ? 1.0f : (float)bc) + SMALL_NUMBER;
    out[i] = out[i] / div;
}

// ---------------------------------------------------------------------------
extern "C" void kernel_launch(void* const* d_in, const int* in_sizes, int n_in,
                              void* d_out, int out_size, void* d_ws, size_t ws_size,
                              hipStream_t stream) {
    const float* ns    = (const float*)d_in[0];  // node_states [100000,256]
    const float* W     = (const float*)d_in[1];  // W  [1536,256]
    const float* b     = (const float*)d_in[2];  // b  [1536]
    const float* Wp    = (const float*)d_in[3];  // Wp [256,256]
    const float* bp    = (const float*)d_in[4];  // bp [256]
    const int*   edges = (const int*)d_in[5];    // [6,400000,2]
    const int*   pos   = (const int*)d_in[6];    // [6,400000]
    float*       out   = (float*)d_out;          // [100000,256]

    char* ws = (char*)d_ws;
    const size_t prop_bytes   = (size_t)N_NODES * DIM * sizeof(float);     // 102.4 MB
    const size_t gating_bytes = (size_t)POS_TABLE * DIM * sizeof(float);   // 0.5 MB
    float* prop_ws   = (float*)ws;
    float* gating_ws = (float*)(ws + prop_bytes);
    int*   bincount  = (int*)(ws + prop_bytes + gating_bytes);

    k_zero<<<2048, 256, 0, stream>>>(out, bincount);
    k_gating<<<POS_TABLE, DIM, 0, stream>>>(Wp, bp, gating_ws);

    for (int i = 0; i < N_EDGE_TYPES; ++i) {
        // prop slice for this edge type (stream order serializes gemm->scatter)
        k_gemm<<<N_NODES / 32, 256, 0, stream>>>(ns, W, b, prop_ws, i);
        k_scatter<<<M_PER_TYPE / 4, 256, 0, stream>>>(
            edges + (size_t)i * M_PER_TYPE * 2,
            pos   + (size_t)i * M_PER_TYPE,
            prop_ws, gating_ws, out, bincount);
    }

    const long long total = (long long)N_NODES * DIM;
    k_div<<<(unsigned)((total + 255) / 256), 256, 0, stream>>>(out, bincount);
}